// Encoder_omics_86569360818307
// MI455X (gfx1250) — compile-verified
//
#include <hip/hip_runtime.h>
#include <hip/hip_bf16.h>

typedef __attribute__((ext_vector_type(2))) float v2f;
typedef __attribute__((ext_vector_type(8))) float v8f;

#define NN 50000
#define EE 800000

// ---------------------------------------------------------------------------
// Zero-init kernel (atomic-accumulation targets must start at 0 every call)
// ---------------------------------------------------------------------------
__global__ void zero_f32(float* p, long n) {
  long i = (long)blockIdx.x * blockDim.x + threadIdx.x;
  if (i < n) p[i] = 0.0f;
}

// ---------------------------------------------------------------------------
// Dense GEMM C[M,N] = A[M,K] @ B[K,N] via V_WMMA_F32_16X16X4_F32.
// K, N are compile-time: B/C strides fold into global_load/store immediate
// offsets, killing the per-load v_add_nc_u64 chains seen with runtime N.
// One wave computes a 16x64 output slab: 4 accumulators share one A fragment
// per k-step -> 4 back-to-back WMMAs (no D->A/B hazard, good XDL pipelining).
// When N % 64 != 0, OOB columns clamp their B pointer (WMMA is column-
// separable so garbage stays in columns we never store); when N % 64 == 0
// all clamp/guard code vanishes at compile time.
// ---------------------------------------------------------------------------
template <int K, int N>
__global__ void gemm_wmma_f32(const float* __restrict__ A,
                              const float* __restrict__ B,
                              float* __restrict__ C, int M) {
  constexpr int nGroups = (N + 63) / 64;
  constexpr bool fullN = (N % 64) == 0;

  const int lane = threadIdx.x & 31;
  const int waveInBlk = threadIdx.x >> 5;
  const int wavesPerBlk = blockDim.x >> 5;
  const int mTiles = M >> 4;
  const long work = (long)blockIdx.x * wavesPerBlk + waveInBlk;
  if (work >= (long)mTiles * nGroups) return;

  const int mt = (int)(work / nGroups);
  const int ng = (int)(work % nGroups);
  const int mBase = mt << 4;

  const int m    = lane & 15;            // A row within tile
  const int kOff = (lane >> 4) << 1;     // K sub-pair: 0 or 2
  const int colBase = (ng << 6) + (lane & 15);

  // A fragment pointer: lane reads (a.x, a.y) = A[row][k+kOff .. k+kOff+1]
  const float* aPtr = A + (long)(mBase + m) * K + kOff;

  const int c0 = colBase, c1 = colBase + 16, c2 = colBase + 32, c3 = colBase + 48;
  int l0 = c0, l1 = c1, l2 = c2, l3 = c3;
  if constexpr (!fullN) {
    l0 = (c0 < N) ? c0 : (N - 1);
    l1 = (c1 < N) ? c1 : (N - 1);
    l2 = (c2 < N) ? c2 : (N - 1);
    l3 = (c3 < N) ? c3 : (N - 1);
  }

  const float* b0 = B + (long)kOff * N + l0;
  const float* b1 = B + (long)kOff * N + l1;
  const float* b2 = B + (long)kOff * N + l2;
  const float* b3 = B + (long)kOff * N + l3;
  constexpr long bStep = 4L * N;         // compile-time stride

  v8f acc0 = {}, acc1 = {}, acc2 = {}, acc3 = {};

  for (int k = 0; k < K; k += 4) {
    v2f a;
    a.x = aPtr[0];
    a.y = aPtr[1];
    __builtin_prefetch(aPtr + 32, 0, 3);   // global_prefetch_b8: 2 lines ahead

    v2f f0, f1, f2, f3;
    f0.x = b0[0]; f0.y = b0[N];
    f1.x = b1[0]; f1.y = b1[N];
    f2.x = b2[0]; f2.y = b2[N];
    f3.x = b3[0]; f3.y = b3[N];

    aPtr += 4;
    b0 += bStep; b1 += bStep; b2 += bStep; b3 += bStep;

    acc0 = __builtin_amdgcn_wmma_f32_16x16x4_f32(false, a, false, f0, (short)0, acc0, false, false);
    acc1 = __builtin_amdgcn_wmma_f32_16x16x4_f32(false, a, false, f1, (short)0, acc1, false, false);
    acc2 = __builtin_amdgcn_wmma_f32_16x16x4_f32(false, a, false, f2, (short)0, acc2, false, false);
    acc3 = __builtin_amdgcn_wmma_f32_16x16x4_f32(false, a, false, f3, (short)0, acc3, false, false);
  }

  // C/D layout: VGPR i -> row (mBase + i) for lanes 0-15, (mBase + i + 8) else
  const int rBase = mBase + ((lane >> 4) ? 8 : 0);
  float* cRow = C + (long)rBase * N;
  if (fullN || c0 < N) {
    #pragma unroll
    for (int i = 0; i < 8; ++i) cRow[(long)i * N + c0] = acc0[i];
  }
  if (fullN || c1 < N) {
    #pragma unroll
    for (int i = 0; i < 8; ++i) cRow[(long)i * N + c1] = acc1[i];
  }
  if (fullN || c2 < N) {
    #pragma unroll
    for (int i = 0; i < 8; ++i) cRow[(long)i * N + c2] = acc2[i];
  }
  if (fullN || c3 < N) {
    #pragma unroll
    for (int i = 0; i < 8; ++i) cRow[(long)i * N + c3] = acc3[i];
  }
}

// ---------------------------------------------------------------------------
// SpMM scatter, d=64: out[rows[e], :] += w[e] * x[cols[e], :]
// One thread per (edge, float4 chunk): b128 gathers, 4x fewer index loads
// than per-dim threading; scatter via global_atomic_add_f32.
// ---------------------------------------------------------------------------
__global__ void spmm64_scatter(const int* __restrict__ rows,
                               const int* __restrict__ cols,
                               const float* __restrict__ w,
                               const float* __restrict__ x,
                               float* __restrict__ out) {
  long idx = (long)blockIdx.x * blockDim.x + threadIdx.x;
  if (idx >= (long)EE * 16) return;
  int e = (int)(idx >> 4);
  int q = (int)(idx & 15);                      // 4-float chunk index
  float4 v = ((const float4*)x)[(long)cols[e] * 16 + q];
  float we = w[e];
  float* o = out + (long)rows[e] * 64 + q * 4;
  atomicAdd(o + 0, we * v.x);
  atomicAdd(o + 1, we * v.y);
  atomicAdd(o + 2, we * v.z);
  atomicAdd(o + 3, we * v.w);
}

// ---------------------------------------------------------------------------
// Attention fusion: one wave per node (wave32; each lane owns dims j, j+32).
//   v = tanh(e @ w_omega); vu = v @ u_omega; alpha = softmax(vu);
//   emb = alpha1*e1 + alpha2*e2
// ---------------------------------------------------------------------------
__global__ void attn_fuse(const float* __restrict__ e1,
                          const float* __restrict__ e2,
                          const float* __restrict__ w_omega,
                          const float* __restrict__ u_omega,
                          float* __restrict__ emb,
                          float* __restrict__ alpha) {
  const long node = ((long)blockIdx.x * blockDim.x + threadIdx.x) >> 5;
  const int lane = threadIdx.x & 31;
  if (node >= NN) return;

  const float* r1 = e1 + node * 64;
  const float* r2 = e2 + node * 64;
  const int j0 = lane, j1 = lane + 32;

  float v10 = 0.f, v11 = 0.f, v20 = 0.f, v21 = 0.f;
  for (int k = 0; k < 64; ++k) {
    float a1 = r1[k], a2 = r2[k];
    float w0 = w_omega[k * 64 + j0];
    float w1 = w_omega[k * 64 + j1];
    v10 = fmaf(a1, w0, v10); v11 = fmaf(a1, w1, v11);
    v20 = fmaf(a2, w0, v20); v21 = fmaf(a2, w1, v21);
  }
  v10 = tanhf(v10); v11 = tanhf(v11);
  v20 = tanhf(v20); v21 = tanhf(v21);

  float p1 = v10 * u_omega[j0] + v11 * u_omega[j1];
  float p2 = v20 * u_omega[j0] + v21 * u_omega[j1];
  #pragma unroll
  for (int off = 16; off > 0; off >>= 1) {
    p1 += __shfl_xor(p1, off, 32);
    p2 += __shfl_xor(p2, off, 32);
  }

  float mx = fmaxf(p1, p2);
  float x1 = __expf(p1 - mx), x2 = __expf(p2 - mx);
  float inv = 1.0f / (x1 + x2);
  float a1 = x1 * inv, a2 = x2 * inv;

  emb[node * 64 + j0] = a1 * r1[j0] + a2 * r2[j0];
  emb[node * 64 + j1] = a1 * r1[j1] + a2 * r2[j1];
  if (lane == 0) {
    alpha[node * 2 + 0] = a1;
    alpha[node * 2 + 1] = a2;
  }
}

// ---------------------------------------------------------------------------
template <int K, int N>
static inline void launch_gemm(const float* A, const float* B, float* C,
                               int M, hipStream_t s) {
  const int mTiles = M / 16;
  const int nGroups = (N + 63) / 64;
  const long work = (long)mTiles * nGroups;
  const int wavesPerBlk = 8;  // 256 threads
  const int grid = (int)((work + wavesPerBlk - 1) / wavesPerBlk);
  gemm_wmma_f32<K, N><<<grid, 256, 0, s>>>(A, B, C, M);
}

extern "C" void kernel_launch(void* const* d_in, const int* in_sizes, int n_in,
                              void* d_out, int out_size, void* d_ws, size_t ws_size,
                              hipStream_t stream) {
  const float* omics1  = (const float*)d_in[0];
  const float* omics2  = (const float*)d_in[1];
  const int*   ei1     = (const int*)d_in[2];   // (2, E): rows then cols
  const float* ew1     = (const float*)d_in[3];
  const int*   ei2     = (const int*)d_in[4];
  const float* ew2     = (const float*)d_in[5];
  const float* W_enc1  = (const float*)d_in[6];
  const float* W_enc2  = (const float*)d_in[7];
  const float* w_omega = (const float*)d_in[8];
  const float* u_omega = (const float*)d_in[9];
  const float* W_dec1  = (const float*)d_in[10];
  const float* W_dec2  = (const float*)d_in[11];

  const int* rows1 = ei1;
  const int* cols1 = ei1 + EE;
  const int* rows2 = ei2;
  const int* cols2 = ei2 + EE;

  // Output layout (flat f32, return order): e1, e2, emb, alpha, d1, d2
  float* out   = (float*)d_out;
  float* e1    = out;
  float* e2    = e1    + (long)NN * 64;
  float* emb   = e2    + (long)NN * 64;
  float* alpha = emb   + (long)NN * 64;
  float* d1    = alpha + (long)NN * 2;
  float* d2    = d1    + (long)NN * 2000;

  // Workspace: x1, x2 (encoder GEMM outputs), t1, t2 (decoder spmm outputs)
  float* x1 = (float*)d_ws;
  float* x2 = x1 + (long)NN * 64;
  float* t1 = x2 + (long)NN * 64;
  float* t2 = t1 + (long)NN * 64;

  // 1) zero atomic-scatter destinations (e1,e2 contiguous; t1,t2 contiguous)
  {
    long n = (long)NN * 64 * 2;
    int grid = (int)((n + 255) / 256);
    zero_f32<<<grid, 256, 0, stream>>>(e1, n);
    zero_f32<<<grid, 256, 0, stream>>>(t1, n);
  }

  // 2) encoder GEMMs (WMMA): x = omics @ W_enc
  launch_gemm<2000, 64>(omics1, W_enc1, x1, NN, stream);
  launch_gemm<500, 64>(omics2, W_enc2, x2, NN, stream);

  // 3) encoder SpMM scatter: e = A @ x
  {
    long n = (long)EE * 16;
    int grid = (int)((n + 255) / 256);
    spmm64_scatter<<<grid, 256, 0, stream>>>(rows1, cols1, ew1, x1, e1);
    spmm64_scatter<<<grid, 256, 0, stream>>>(rows2, cols2, ew2, x2, e2);
  }

  // 4) attention fusion -> emb, alpha
  {
    int wavesPerBlk = 8;  // 256 threads
    int grid = (NN + wavesPerBlk - 1) / wavesPerBlk;
    attn_fuse<<<grid, 256, 0, stream>>>(e1, e2, w_omega, u_omega, emb, alpha);
  }

  // 5) decoder reassociated: t = A @ emb (64-wide scatter, ~30x less traffic
  //    than scattering the 2000-wide decoded rows)
  {
    long n = (long)EE * 16;
    int grid = (int)((n + 255) / 256);
    spmm64_scatter<<<grid, 256, 0, stream>>>(rows1, cols1, ew1, emb, t1);
    spmm64_scatter<<<grid, 256, 0, stream>>>(rows2, cols2, ew2, emb, t2);
  }

  // 6) decoder GEMMs (WMMA): d = t @ W_dec
  launch_gemm<64, 2000>(t1, W_dec1, d1, NN, stream);
  launch_gemm<64, 500>(t2, W_dec2, d2, NN, stream);
}